// GNNStack_2602750182101
// MI455X (gfx1250) — compile-verified
//
#include <hip/hip_runtime.h>
#include <hip/hip_bf16.h>
#include <math.h>

typedef __attribute__((ext_vector_type(2))) float v2f;
typedef __attribute__((ext_vector_type(8))) float v8f;

// ---------------------------------------------------------------------------
// init: xbuf = x_in ; acc = 0 ; bias = 0     (n = N*8 elements)
// ---------------------------------------------------------------------------
__global__ void gnn_init_kernel(const float* __restrict__ xin,
                                float* __restrict__ xbuf,
                                float* __restrict__ acc,
                                float* __restrict__ bias, int n) {
  int i = blockIdx.x * blockDim.x + threadIdx.x;
  if (i < n) {
    xbuf[i] = xin[i];
    acc[i]  = 0.0f;
    bias[i] = 0.0f;
  }
}

// ---------------------------------------------------------------------------
// Hoisted loop-invariant: bias[dst[e], i] += max_j edge_attr[e, i, j]
// One thread per (e,i) row of 8 contiguous floats. Streams 410 MB once.
// ---------------------------------------------------------------------------
__global__ void gnn_edge_bias_kernel(const float* __restrict__ edge_attr,
                                     const int* __restrict__ dst,
                                     float* __restrict__ bias,
                                     long long rows /* = E*8 */) {
  long long r = (long long)blockIdx.x * blockDim.x + threadIdx.x;
  if (r >= rows) return;
  const float* p = edge_attr + r * 8;
  __builtin_prefetch(p + 8 * 512, 0, 0);   // stream-ahead -> global_prefetch_b8
  const float4 a = ((const float4*)p)[0];
  const float4 b = ((const float4*)p)[1];
  float m = fmaxf(fmaxf(fmaxf(a.x, a.y), fmaxf(a.z, a.w)),
                  fmaxf(fmaxf(b.x, b.y), fmaxf(b.z, b.w)));
  long long e = r >> 3;
  int i = (int)(r & 7);
  atomicAdd(&bias[(long long)dst[e] * 8 + i], m);
}

// ---------------------------------------------------------------------------
// SpMM step: acc[dst[e]] += x[src[e]]   (one thread per edge, 8 f32 atomics)
// x and acc are 1.6 MB -> L2 resident.
// ---------------------------------------------------------------------------
__global__ void gnn_scatter_kernel(const float* __restrict__ xin,
                                   const int* __restrict__ src,
                                   const int* __restrict__ dst,
                                   float* __restrict__ acc, int E) {
  int e = blockIdx.x * blockDim.x + threadIdx.x;
  if (e >= E) return;
  int s = src[e];
  int d = dst[e];
  const float4 lo = ((const float4*)(xin + (size_t)s * 8))[0];
  const float4 hi = ((const float4*)(xin + (size_t)s * 8))[1];
  float* ap = acc + (size_t)d * 8;
  atomicAdd(ap + 0, lo.x); atomicAdd(ap + 1, lo.y);
  atomicAdd(ap + 2, lo.z); atomicAdd(ap + 3, lo.w);
  atomicAdd(ap + 4, hi.x); atomicAdd(ap + 5, hi.y);
  atomicAdd(ap + 6, hi.z); atomicAdd(ap + 7, hi.w);
}

// ---------------------------------------------------------------------------
// Mean-centering as WMMA:  x = (acc + bias) * M,  M = I - J/8  (8x8),
// computed as D(16x16) = A(16x4)*B(4x16) + D, twice (K=0..3, K=4..7).
// Each wave handles 16 nodes. Also zeroes acc for the next layer's atomics.
//
// f32 layouts (ISA 7.12.2):
//   A 16x4 : lanes 0-15 -> M=lane (lanes 16-31 repeat M);
//            VGPR0 = K=2*hi, VGPR1 = K=2*hi+1   (hi = lane>=16)
//   B 4x16 : lanes 0-15 col=lane, rows 2*hi / 2*hi+1 in VGPR0/1
//   D 16x16: VGPR v -> row v (lanes 0-15, col=lane) / row v+8 (lanes 16-31)
// ---------------------------------------------------------------------------
__device__ __forceinline__ float center_m(int k, int c) {
  if (c >= 8) return 0.0f;                     // zero-pad cols 8..15
  return (k == c) ? 0.875f : -0.125f;          // I - J/8
}

__global__ void gnn_center_wmma_kernel(const float* __restrict__ acc,
                                       const float* __restrict__ bias,
                                       float* __restrict__ xout,
                                       float* __restrict__ acc_zero,
                                       int N) {
  const int lane = threadIdx.x & 31;
  const int wave = threadIdx.x >> 5;
  const int base = (blockIdx.x * (blockDim.x >> 5) + wave) * 16;  // 16 nodes/wave
  if (base >= N) return;            // wave-uniform guard: EXEC all-ones for WMMA

  const int mrow = lane & 15;       // A-row / B-col / D-col index
  const int hi   = lane >> 4;       // K-half selector
  const int k0   = hi * 2;

  const float* ar = acc  + (size_t)(base + mrow) * 8;
  const float* br = bias + (size_t)(base + mrow) * 8;

  v2f a0, a1;                       // A tiles: t(node, k) = acc + bias
  a0.x = ar[k0 + 0] + br[k0 + 0];
  a0.y = ar[k0 + 1] + br[k0 + 1];
  a1.x = ar[k0 + 4] + br[k0 + 4];
  a1.y = ar[k0 + 5] + br[k0 + 5];

  v2f b0, b1;                       // B tiles: centering matrix rows
  b0.x = center_m(k0 + 0, mrow);
  b0.y = center_m(k0 + 1, mrow);
  b1.x = center_m(k0 + 4, mrow);
  b1.y = center_m(k0 + 5, mrow);

  v8f c = {};
  c = __builtin_amdgcn_wmma_f32_16x16x4_f32(false, a0, false, b0,
                                            (short)0, c, false, false);
  c = __builtin_amdgcn_wmma_f32_16x16x4_f32(false, a1, false, b1,
                                            (short)0, c, false, false);

  // D cols 0..7 are the centered features; lanes with mrow>=8 hold zero cols.
  if (mrow < 8) {
    const int nodeBase = base + hi * 8;   // rows v (lanes<16) / v+8 (lanes>=16)
#pragma unroll
    for (int v = 0; v < 8; ++v) {
      xout[(size_t)(nodeBase + v) * 8 + mrow] = c[v];
    }
  }

  // Re-zero this wave's 128-float accumulator slab for the next layer.
  float4 z; z.x = 0.f; z.y = 0.f; z.z = 0.f; z.w = 0.f;
  ((float4*)(acc_zero + (size_t)base * 8))[lane] = z;
}

// ---------------------------------------------------------------------------
// Final: out = acc ; out2 = log_softmax(acc, axis=1). One thread per node.
// ---------------------------------------------------------------------------
__global__ void gnn_final_kernel(const float* __restrict__ acc,
                                 float* __restrict__ out, int N) {
  int n = blockIdx.x * blockDim.x + threadIdx.x;
  if (n >= N) return;
  const float* a = acc + (size_t)n * 8;
  float v[8];
  float m = -INFINITY;
#pragma unroll
  for (int i = 0; i < 8; ++i) { v[i] = a[i]; m = fmaxf(m, v[i]); }
  float s = 0.0f;
#pragma unroll
  for (int i = 0; i < 8; ++i) s += __expf(v[i] - m);
  float lse = m + __logf(s);
  float* o0 = out + (size_t)n * 8;
  float* o1 = out + (size_t)N * 8 + (size_t)n * 8;
#pragma unroll
  for (int i = 0; i < 8; ++i) { o0[i] = v[i]; o1[i] = v[i] - lse; }
}

// ---------------------------------------------------------------------------
extern "C" void kernel_launch(void* const* d_in, const int* in_sizes, int n_in,
                              void* d_out, int out_size, void* d_ws, size_t ws_size,
                              hipStream_t stream) {
  const float* x_in      = (const float*)d_in[0];
  const float* edge_attr = (const float*)d_in[1];
  const int*   eidx      = (const int*)d_in[2];

  const int N = in_sizes[0] / 8;       // 50000
  const int E = in_sizes[2] / 2;       // 1600000
  const int* src = eidx;
  const int* dst = eidx + E;

  float* xbuf = (float*)d_ws;                       // N*8
  float* acc  = xbuf + (size_t)N * 8;               // N*8
  float* bias = acc  + (size_t)N * 8;               // N*8

  const int NB_NODEF = (N * 8 + 255) / 256;
  gnn_init_kernel<<<NB_NODEF, 256, 0, stream>>>(x_in, xbuf, acc, bias, N * 8);

  const long long rows = (long long)E * 8;
  const int NB_ROWS = (int)((rows + 255) / 256);
  gnn_edge_bias_kernel<<<NB_ROWS, 256, 0, stream>>>(edge_attr, dst, bias, rows);

  const int NB_E = (E + 255) / 256;
  const int NB_C = (N + 127) / 128;    // 8 waves/block * 16 nodes/wave
  for (int layer = 0; layer < 29; ++layer) {
    gnn_scatter_kernel<<<NB_E, 256, 0, stream>>>(xbuf, src, dst, acc, E);
    gnn_center_wmma_kernel<<<NB_C, 256, 0, stream>>>(acc, bias, xbuf, acc, N);
  }

  // Final layer: plain segment_sum of x[src] (acc was zeroed by last center).
  gnn_scatter_kernel<<<NB_E, 256, 0, stream>>>(xbuf, src, dst, acc, E);
  gnn_final_kernel<<<(N + 255) / 256, 256, 0, stream>>>(acc, (float*)d_out, N);
}